// FastMaskedConv2D_60430189854965
// MI455X (gfx1250) — compile-verified
//
#include <hip/hip_runtime.h>

// FastMaskedConv2D incremental step as a tiled f32 GEMM on CDNA5 WMMA.
//   y[b,f] = sum_{h,w,ci} window[b,h,w,ci] * masked_kernel[h,w,ci,f] + bias[f]
// window column c = iw + w - 4 (zero-padded); (h=4, c=iw) line replaced by
// `inputs`; whole cache shifted one H-row when iw==0; (h=4, w>=5) taps and
// out-of-range columns contribute nothing and are skipped at runtime.
// Pipeline: double-buffered GLOBAL_LOAD_ASYNC_TO_LDS_B128 (ASYNCcnt) feeding
// V_WMMA_F32_16X16X4_F32 accumulation chains.

typedef float v2f __attribute__((ext_vector_type(2)));
typedef float v4f __attribute__((ext_vector_type(4)));
typedef float v8f __attribute__((ext_vector_type(8)));
typedef int   v4i __attribute__((ext_vector_type(4)));

// Address-space-qualified int4 pointers for the async-to-LDS builtin
// (prototype: void(v4i AS1*, v4i AS3*, Imm int offset, Imm int cpol)).
typedef __attribute__((address_space(1))) v4i gv4i;
typedef __attribute__((address_space(3))) v4i lv4i;

constexpr int kL     = 64;
constexpr int kKH    = 5;
constexpr int kKW    = 9;
constexpr int kFEAT  = 64;
constexpr int kBATCH = 4096;
constexpr int kCUT   = kKW / 2 + 1;   // EXCLUSIVE=False -> 5

constexpr int MTILE  = 16;            // batch rows per block -> 256 blocks
constexpr int ASTR   = 68;            // LDS stride (floats): 272B rows, 16B aligned, conflict-free
constexpr int BSTR   = 72;            // LDS stride (floats): 288B rows, 16B aligned, conflict-free
constexpr int MAXTAP = kKH * kKW;

#if defined(__has_builtin)
#  if __has_builtin(__builtin_amdgcn_global_load_async_to_lds_b128) && \
      __has_builtin(__builtin_amdgcn_s_wait_asynccnt)
#    define USE_ASYNC 1
#  endif
#endif
#ifndef USE_ASYNC
#  define USE_ASYNC 0
#endif

// One 16-byte memory->LDS copy per lane. Async path: no VGPR staging,
// tracked by ASYNCcnt. Fallback: plain vector copy.
__device__ __forceinline__ void copy16_to_lds(const float* g, float* l) {
#if USE_ASYNC
  __builtin_amdgcn_global_load_async_to_lds_b128(
      (gv4i*)(uintptr_t)g,
      (lv4i*)(uint32_t)(uintptr_t)l,
      0, 0);
#else
  *(v4f*)l = *(const v4f*)g;
#endif
}

__global__ __launch_bounds__(128)
void fastmaskedconv_wmma_kernel(const float* __restrict__ inputs,
                                const float* __restrict__ cache,
                                const float* __restrict__ kern,
                                const float* __restrict__ bias,
                                const int*   __restrict__ indexp,
                                float*       __restrict__ out) {
  __shared__ float As[2][MTILE * ASTR];   // 2 x 16 rows x 64 ci
  __shared__ float Bs[2][kFEAT * BSTR];   // 2 x 64 ci  x 64 f
  __shared__ int   taps[MAXTAP];
  __shared__ int   ntaps;

  const int tid   = threadIdx.x;
  const int lane  = tid & 31;
  const int wave  = tid >> 5;          // 0..3 -> N-tile
  const int lhalf = lane >> 4;
  const int l16   = lane & 15;
  const int mbase = blockIdx.x * MTILE;
  const int n0    = wave * 16;

  // ---- build uniform active-tap list (runtime index logic) ----
  if (tid == 0) {
    const int idx = indexp[0];
    int iw = idx % kL; if (iw < 0) iw += kL;        // python-style mod
    const bool do_update = (idx >= 0);
    const bool shifted   = do_update && (iw == 0);
    int n = 0;
    for (int h = 0; h < kKH; ++h) {
      const int wlim = (h == kKH - 1) ? kCUT : kKW; // masked taps skipped
      for (int w = 0; w < wlim; ++w) {
        const int c = iw + w - kKW / 2;             // cache column
        if (c < 0 || c >= kL) continue;             // zero pad -> skip
        const bool rep = do_update && (h == kKH - 1) && (c == iw);
        if (shifted && (h == kKH - 1) && !rep) continue; // shifted-in zeros
        const int hh = shifted ? (h + 1) : h;       // unused when rep
        taps[n++] = c | (hh << 8) | ((h * kKW + w) << 12) | (rep ? (1 << 18) : 0);
      }
    }
    ntaps = n;
  }
  __syncthreads();
  const int nt = ntaps;

  // ---- per-tap async stage: 10 x b128 per lane (A:2, B:8) ----
  auto issue = [&](int ti, int b) {
    const int  code = taps[ti];
    const int  c    = code & 0xFF;
    const int  hh   = (code >> 8) & 0xF;
    const int  ki   = (code >> 12) & 0x3F;
    const bool rep  = (code >> 18) & 1;
    #pragma unroll
    for (int i = 0; i < 2; ++i) {                  // A tile: 16 x 64 f32
      const int sl  = tid * 2 + i;                 // 0..255 float4 slots
      const int row = sl >> 4;
      const int c4  = sl & 15;
      const float* src = rep
        ? inputs + (size_t)(mbase + row) * kFEAT + c4 * 4
        : cache + (((size_t)(mbase + row) * kKH + hh) * kL + c) * kFEAT + c4 * 4;
      copy16_to_lds(src, &As[b][row * ASTR + c4 * 4]);
    }
    const float* kb = kern + (size_t)ki * kFEAT * kFEAT;
    #pragma unroll
    for (int i = 0; i < 8; ++i) {                  // B tile: 64 x 64 f32
      const int sl  = tid * 8 + i;                 // 0..1023 float4 slots
      const int row = sl >> 4;
      const int c4  = sl & 15;
      copy16_to_lds(kb + row * kFEAT + c4 * 4, &Bs[b][row * BSTR + c4 * 4]);
    }
  };

  // ---- double-buffered pipeline: prefetch tap i+1 while computing tap i ----
  v8f acc = {};
  int buf = 0;
  if (nt > 0) issue(0, 0);
  for (int i = 0; i < nt; ++i) {
    if (i + 1 < nt) {
      issue(i + 1, buf ^ 1);
#if USE_ASYNC
      __builtin_amdgcn_s_wait_asynccnt(10);  // in-order -> tap i's 10 ops done
#endif
    } else {
#if USE_ASYNC
      __builtin_amdgcn_s_wait_asynccnt(0);
#endif
    }
    __syncthreads();                         // all waves' quarters landed

    const float* Ab = As[buf];
    const float* Bb = Bs[buf];
    #pragma unroll
    for (int k0 = 0; k0 < kFEAT; k0 += 4) {  // 16 x V_WMMA_F32_16X16X4_F32
      const int kk = k0 + 2 * lhalf;         // this half-wave's K pair
      v2f a = *(const v2f*)&Ab[l16 * ASTR + kk];   // A: VGPR0=K{0,2}, VGPR1=K{1,3}
      v2f bfrag;                                    // B: N across lanes
      bfrag.x = Bb[kk * BSTR + n0 + l16];
      bfrag.y = Bb[(kk + 1) * BSTR + n0 + l16];
      acc = __builtin_amdgcn_wmma_f32_16x16x4_f32(false, a, false, bfrag,
                                                  (short)0, acc, false, false);
    }
    __syncthreads();                         // readers done before buf reuse
    buf ^= 1;
  }

  // ---- epilogue: bias + store (C layout: M = i + 8*lhalf, N = l16) ----
  const float bv = bias[n0 + l16];
  #pragma unroll
  for (int i = 0; i < 8; ++i) {
    const int row = mbase + i + 8 * lhalf;
    out[(size_t)row * kFEAT + n0 + l16] = acc[i] + bv;
  }
}

extern "C" void kernel_launch(void* const* d_in, const int* in_sizes, int n_in,
                              void* d_out, int out_size, void* d_ws, size_t ws_size,
                              hipStream_t stream) {
  const float* inputs = (const float*)d_in[0];   // (4096, 64)
  const float* cache  = (const float*)d_in[1];   // (4096, 5, 64, 64)
  const float* kern   = (const float*)d_in[2];   // (5, 9, 64, 64)
  const float* bias   = (const float*)d_in[3];   // (64,)
  const int*   index  = (const int*)d_in[4];     // scalar on device
  float* out = (float*)d_out;                    // (4096, 64)

  dim3 grid(kBATCH / MTILE);   // 256 blocks
  dim3 block(128);             // 4 waves (wave32)
  hipLaunchKernelGGL(fastmaskedconv_wmma_kernel, grid, block, 0, stream,
                     inputs, cache, kern, bias, index, out);
}